// V2FNSAttention_22162031247485
// MI455X (gfx1250) — compile-verified
//
#include <hip/hip_runtime.h>
#include <math.h>

typedef float v2f __attribute__((ext_vector_type(2)));
typedef float v8f __attribute__((ext_vector_type(8)));

#define S_LEN   2048
#define EDIM    768
#define NHEADS  12
#define HD      64
#define NBH     24      // B * NHEADS
#define NROWS   4096    // B * S

// ---------------------------------------------------------------------------
// GEMM: out = X @ W^T + bias (+ optional residual).  X:[NROWS,EDIM] W:[EDIM,EDIM]
// Block tile M=32, N=128, K-step 32, double-buffered LDS.
// 256 threads = 8 waves; wave computes 16 rows x 32 cols (2 C tiles, shared A).
// mode 0: write head layout   mode 1: row-major + residual
// ---------------------------------------------------------------------------
__global__ __launch_bounds__(256) void proj_kernel(const float* __restrict__ X,
    const float* __restrict__ W, const float* __restrict__ bias,
    float* __restrict__ out, const float* __restrict__ resid, int mode)
{
    __shared__ float Xs[2][32][33];     // padded A-frag reads
    __shared__ float Ws[2][32][128];    // Ws[k][n] = W[n0+n][kk+k]

    const int tid  = threadIdx.x;
    const int w    = tid >> 5, lane = tid & 31;
    const int wm   = w & 1,    wcg  = w >> 1;          // col group of 32
    const int m0   = blockIdx.y * 32;
    const int n0   = blockIdx.x * 128;
    const int ln   = lane & 15;
    const int half = lane >> 4;
    const int koff = half << 1;

    // cooperative-load coordinates
    const int xr = (tid * 4) >> 5, xk = (tid * 4) & 31;          // X: float4 per thread
    int wn_[4], wk_[4];                                           // W: 4 float4 per thread
    #pragma unroll
    for (int e = 0; e < 4; ++e) {
        int idx = tid * 4 + e;
        wn_[e] = idx >> 3;           // 0..127
        wk_[e] = (idx & 7) * 4;      // 0..28
    }

    v8f c0 = {}, c1 = {};

    // prologue: stage k-chunk 0 into buffer 0
    {
        float4 xv = *(const float4*)(X + (size_t)(m0 + xr) * EDIM + xk);
        Xs[0][xr][xk] = xv.x; Xs[0][xr][xk+1] = xv.y;
        Xs[0][xr][xk+2] = xv.z; Xs[0][xr][xk+3] = xv.w;
        #pragma unroll
        for (int e = 0; e < 4; ++e) {
            float4 wv = *(const float4*)(W + (size_t)(n0 + wn_[e]) * EDIM + wk_[e]);
            Ws[0][wk_[e]  ][wn_[e]] = wv.x;
            Ws[0][wk_[e]+1][wn_[e]] = wv.y;
            Ws[0][wk_[e]+2][wn_[e]] = wv.z;
            Ws[0][wk_[e]+3][wn_[e]] = wv.w;
        }
    }

    int buf = 0;
    for (int kk = 0; kk < EDIM; kk += 32) {
        __syncthreads();
        const bool has = (kk + 32 < EDIM);
        float4 xpre; float4 wpre[4];
        if (has) {   // issue next-chunk global loads early; wait lands after WMMAs
            xpre = *(const float4*)(X + (size_t)(m0 + xr) * EDIM + kk + 32 + xk);
            #pragma unroll
            for (int e = 0; e < 4; ++e)
                wpre[e] = *(const float4*)(W + (size_t)(n0 + wn_[e]) * EDIM + kk + 32 + wk_[e]);
        }
        // batched fragment loads (one DS clause), then WMMA chain
        v2f a[8], b0[8], b1[8];
        #pragma unroll
        for (int s = 0; s < 8; ++s) {
            int k0 = s * 4 + koff;
            a[s].x  = Xs[buf][wm * 16 + ln][k0];
            a[s].y  = Xs[buf][wm * 16 + ln][k0 + 1];
            b0[s].x = Ws[buf][k0    ][wcg * 32 + ln];
            b0[s].y = Ws[buf][k0 + 1][wcg * 32 + ln];
            b1[s].x = Ws[buf][k0    ][wcg * 32 + 16 + ln];
            b1[s].y = Ws[buf][k0 + 1][wcg * 32 + 16 + ln];
        }
        #pragma unroll
        for (int s = 0; s < 8; ++s) {
            c0 = __builtin_amdgcn_wmma_f32_16x16x4_f32(false, a[s], false, b0[s], (short)0, c0, false, false);
            c1 = __builtin_amdgcn_wmma_f32_16x16x4_f32(false, a[s], false, b1[s], (short)0, c1, false, false);
        }
        if (has) {
            int nb = buf ^ 1;
            Xs[nb][xr][xk] = xpre.x; Xs[nb][xr][xk+1] = xpre.y;
            Xs[nb][xr][xk+2] = xpre.z; Xs[nb][xr][xk+3] = xpre.w;
            #pragma unroll
            for (int e = 0; e < 4; ++e) {
                Ws[nb][wk_[e]  ][wn_[e]] = wpre[e].x;
                Ws[nb][wk_[e]+1][wn_[e]] = wpre[e].y;
                Ws[nb][wk_[e]+2][wn_[e]] = wpre[e].z;
                Ws[nb][wk_[e]+3][wn_[e]] = wpre[e].w;
            }
        }
        buf ^= 1;
    }

    const int gc0 = n0 + wcg * 32 + ln;
    const int gc1 = gc0 + 16;
    const float bv0 = bias[gc0], bv1 = bias[gc1];
    #pragma unroll
    for (int r = 0; r < 8; ++r) {
        int grow = m0 + wm * 16 + r + (half << 3);
        float v0 = c0[r] + bv0, v1 = c1[r] + bv1;
        if (mode == 1) {
            size_t o0 = (size_t)grow * EDIM + gc0;
            size_t o1 = (size_t)grow * EDIM + gc1;
            out[o0] = v0 + resid[o0];
            out[o1] = v1 + resid[o1];
        } else {
            int b_ = grow >> 11, s_ = grow & 2047;
            int h0 = gc0 >> 6, d0 = gc0 & 63;
            int h1 = gc1 >> 6, d1 = gc1 & 63;
            out[(((size_t)(b_ * NHEADS + h0)) * S_LEN + s_) * HD + d0] = v0;
            out[(((size_t)(b_ * NHEADS + h1)) * S_LEN + s_) * HD + d1] = v1;
        }
    }
}

// ---------------------------------------------------------------------------
// Row squared-norms of q in head layout
// ---------------------------------------------------------------------------
__global__ __launch_bounds__(256) void norm_kernel(const float* __restrict__ q,
                                                   float* __restrict__ qn)
{
    int i = blockIdx.x * blockDim.x + threadIdx.x;
    if (i >= NBH * S_LEN) return;
    const float* p = q + (size_t)i * HD;
    float s = 0.f;
    #pragma unroll
    for (int d = 0; d < HD; ++d) s += p[d] * p[d];
    qn[i] = s;
}

// ---------------------------------------------------------------------------
// Streaming distance-attention, double-buffered j-tiles.
// Grid: (S/128, B*H). 8 waves, wave owns 16 query rows.
// scores in (0,1] => exp(score) in (1,e]: single pass, no max tracking.
// ---------------------------------------------------------------------------
__global__ __launch_bounds__(256) void attn_kernel(const float* __restrict__ q,
    const float* __restrict__ v, const float* __restrict__ qn,
    const float* __restrict__ mask, float* __restrict__ ctx)
{
    __shared__ float Qj[2][16][68];
    __shared__ float Vj[2][16][64];
    __shared__ float Pw[8][16][17];
    __shared__ float qnj[2][16];
    __shared__ float mkj[2][16];

    const int tid  = threadIdx.x;
    const int w    = tid >> 5, lane = tid & 31;
    const int ln   = lane & 15;
    const int half = lane >> 4;
    const int koff = half << 1;
    const int bh   = blockIdx.y;
    const int b_   = bh / NHEADS;
    const int h_   = bh % NHEADS;
    const int i0   = blockIdx.x * 128 + w * 16;

    const float* qbh  = q  + (size_t)bh * S_LEN * HD;
    const float* vbh  = v  + (size_t)bh * S_LEN * HD;
    const float* qnbh = qn + (size_t)bh * S_LEN;
    const float* mk   = mask + (size_t)b_ * S_LEN;

    // cooperative-load coordinates (16x64 tile, float4 per thread)
    const int jr = (tid * 4) >> 6, jd = (tid * 4) & 63;

    // preload A-fragments of q_i (reused across all 128 j-steps)
    v2f afr[16];
    {
        const float* qrow = qbh + (size_t)(i0 + ln) * HD;
        #pragma unroll
        for (int f = 0; f < 16; ++f) {
            afr[f].x = qrow[f * 4 + koff];
            afr[f].y = qrow[f * 4 + koff + 1];
        }
    }
    float qni[8], keepi[8];
    #pragma unroll
    for (int r = 0; r < 8; ++r) {
        int m = r + (half << 3);
        qni[r]   = qnbh[i0 + m];
        keepi[r] = (mk[i0 + m] >= 0.0f) ? 1.0f : 0.0f;
    }

    v8f oacc0 = {}, oacc1 = {}, oacc2 = {}, oacc3 = {};
    v8f rsum = {};

    // prologue: stage tile j0=0 into buffer 0
    {
        float4 qv = *(const float4*)(qbh + (size_t)jr * HD + jd);
        *(float4*)&Qj[0][jr][jd] = qv;
        float4 vv = *(const float4*)(vbh + (size_t)jr * HD + jd);
        *(float4*)&Vj[0][jr][jd] = vv;
        if (tid < 16) { qnj[0][tid] = qnbh[tid]; mkj[0][tid] = mk[tid]; }
    }

    int buf = 0;
    for (int j0 = 0; j0 < S_LEN; j0 += 16) {
        __syncthreads();
        const bool has = (j0 + 16 < S_LEN);
        float4 qpre, vpre; float qnpre = 0.f, mkpre = 0.f;
        if (has) {   // issue next-tile loads before WMMA work
            qpre = *(const float4*)(qbh + (size_t)(j0 + 16 + jr) * HD + jd);
            vpre = *(const float4*)(vbh + (size_t)(j0 + 16 + jr) * HD + jd);
            if (tid < 16) { qnpre = qnbh[j0 + 16 + tid]; mkpre = mk[j0 + 16 + tid]; }
        }

        // gemm1: dot_ij = q_i . q_j   (batched B-frags, 2 accumulator chains)
        v2f bb[16];
        #pragma unroll
        for (int f = 0; f < 16; ++f) {
            bb[f].x = Qj[buf][ln][f * 4 + koff];
            bb[f].y = Qj[buf][ln][f * 4 + koff + 1];
        }
        v8f cc0 = {}, cc1 = {};
        #pragma unroll
        for (int f = 0; f < 16; f += 2) {
            cc0 = __builtin_amdgcn_wmma_f32_16x16x4_f32(false, afr[f],   false, bb[f],   (short)0, cc0, false, false);
            cc1 = __builtin_amdgcn_wmma_f32_16x16x4_f32(false, afr[f+1], false, bb[f+1], (short)0, cc1, false, false);
        }
        v8f cc = cc0 + cc1;

        // scores -> unnormalized exp (TRANS pipe overlaps XDL)
        float qnjl  = qnj[buf][ln];
        float keepj = (mkj[buf][ln] >= 0.0f) ? 1.0f : 0.0f;
        #pragma unroll
        for (int r = 0; r < 8; ++r) {
            float d2   = qni[r] + qnjl - 2.0f * cc[r];
            float dist = sqrtf(fmaxf(d2, 1e-12f));
            dist = (keepi[r] * keepj > 0.0f) ? dist : 2.0f;
            float sc = __powf(1.0f + dist, -3.5f);   // (1+d)^(-D-beta)
            float e  = __expf(sc);
            rsum[r] += e;
            Pw[w][r + (half << 3)][ln] = e;          // C-layout -> LDS transpose
        }
        asm volatile("s_wait_dscnt 0" ::: "memory"); // per-wave LDS RAW fence

        // gemm2: O += P @ V  (batched frags, 4 independent accumulator chains)
        v2f aa[4], vb[16];
        #pragma unroll
        for (int t = 0; t < 4; ++t) {
            int k0 = t * 4 + koff;
            aa[t].x = Pw[w][ln][k0];
            aa[t].y = Pw[w][ln][k0 + 1];
            #pragma unroll
            for (int td = 0; td < 4; ++td) {
                vb[t * 4 + td].x = Vj[buf][k0    ][td * 16 + ln];
                vb[t * 4 + td].y = Vj[buf][k0 + 1][td * 16 + ln];
            }
        }
        #pragma unroll
        for (int t = 0; t < 4; ++t) {
            oacc0 = __builtin_amdgcn_wmma_f32_16x16x4_f32(false, aa[t], false, vb[t*4+0], (short)0, oacc0, false, false);
            oacc1 = __builtin_amdgcn_wmma_f32_16x16x4_f32(false, aa[t], false, vb[t*4+1], (short)0, oacc1, false, false);
            oacc2 = __builtin_amdgcn_wmma_f32_16x16x4_f32(false, aa[t], false, vb[t*4+2], (short)0, oacc2, false, false);
            oacc3 = __builtin_amdgcn_wmma_f32_16x16x4_f32(false, aa[t], false, vb[t*4+3], (short)0, oacc3, false, false);
        }

        if (has) {   // commit prefetched tile into alternate buffer
            int nb = buf ^ 1;
            *(float4*)&Qj[nb][jr][jd] = qpre;
            *(float4*)&Vj[nb][jr][jd] = vpre;
            if (tid < 16) { qnj[nb][tid] = qnpre; mkj[nb][tid] = mkpre; }
        }
        buf ^= 1;
    }

    // reduce row sums across the 16 lanes of each half
    #pragma unroll
    for (int r = 0; r < 8; ++r) {
        float s = rsum[r];
        s += __shfl_xor(s, 1);
        s += __shfl_xor(s, 2);
        s += __shfl_xor(s, 4);
        s += __shfl_xor(s, 8);
        rsum[r] = s;
    }
    // normalize + scatter to context [B,S,E]
    #pragma unroll
    for (int r = 0; r < 8; ++r) {
        int m = r + (half << 3);
        size_t row = (size_t)(b_ * S_LEN + i0 + m) * EDIM + h_ * HD + ln;
        float inv = 1.0f / rsum[r];
        ctx[row +  0] = oacc0[r] * inv;
        ctx[row + 16] = oacc1[r] * inv;
        ctx[row + 32] = oacc2[r] * inv;
        ctx[row + 48] = oacc3[r] * inv;
    }
}

// ---------------------------------------------------------------------------
// LayerNorm over E=768 per row (input already has residual added)
// ---------------------------------------------------------------------------
__global__ __launch_bounds__(256) void ln_kernel(const float* __restrict__ y,
    const float* __restrict__ g, const float* __restrict__ be,
    float* __restrict__ out)
{
    __shared__ float red[256];
    __shared__ float red2[256];
    const int row = blockIdx.x;
    const int t = threadIdx.x;
    const float* yr = y + (size_t)row * EDIM;
    float x0 = yr[t], x1 = yr[t + 256], x2 = yr[t + 512];
    red[t]  = x0 + x1 + x2;
    red2[t] = x0 * x0 + x1 * x1 + x2 * x2;
    __syncthreads();
    for (int off = 128; off > 0; off >>= 1) {
        if (t < off) { red[t] += red[t + off]; red2[t] += red2[t + off]; }
        __syncthreads();
    }
    float mu  = red[0] * (1.0f / EDIM);
    float var = red2[0] * (1.0f / EDIM) - mu * mu;
    float rs  = rsqrtf(var + 1e-12f);
    size_t o = (size_t)row * EDIM;
    out[o + t]       = (x0 - mu) * rs * g[t]       + be[t];
    out[o + t + 256] = (x1 - mu) * rs * g[t + 256] + be[t + 256];
    out[o + t + 512] = (x2 - mu) * rs * g[t + 512] + be[t + 512];
}

// ---------------------------------------------------------------------------
extern "C" void kernel_launch(void* const* d_in, const int* in_sizes, int n_in,
                              void* d_out, int out_size, void* d_ws, size_t ws_size,
                              hipStream_t stream)
{
    const float* x    = (const float*)d_in[0];
    const float* mask = (const float*)d_in[1];
    const float* Wq   = (const float*)d_in[2];
    const float* bq   = (const float*)d_in[3];
    const float* Wv   = (const float*)d_in[4];
    const float* bv   = (const float*)d_in[5];
    const float* Wo   = (const float*)d_in[6];
    const float* bo   = (const float*)d_in[7];
    const float* lng  = (const float*)d_in[8];
    const float* lnb  = (const float*)d_in[9];

    const size_t NQ = (size_t)NBH * S_LEN * HD;
    float* ws   = (float*)d_ws;
    float* q    = ws;
    float* v    = q + NQ;
    float* qn   = v + NQ;
    float* ctx  = qn + (size_t)NBH * S_LEN;
    float* ybuf = ctx + (size_t)NROWS * EDIM;

    dim3 gproj(EDIM / 128, NROWS / 32);             // (6, 128)
    proj_kernel<<<gproj, 256, 0, stream>>>(x, Wq, bq, q, nullptr, 0);
    proj_kernel<<<gproj, 256, 0, stream>>>(x, Wv, bv, v, nullptr, 0);

    norm_kernel<<<(NBH * S_LEN + 255) / 256, 256, 0, stream>>>(q, qn);

    dim3 gattn(S_LEN / 128, NBH);                   // (16, 24)
    attn_kernel<<<gattn, 256, 0, stream>>>(q, v, qn, mask, ctx);

    proj_kernel<<<gproj, 256, 0, stream>>>(ctx, Wo, bo, ybuf, x, 1);

    ln_kernel<<<NROWS, 256, 0, stream>>>(ybuf, lng, lnb, (float*)d_out);
}